// MoogVCFRFModelCell_55035710931671
// MI455X (gfx1250) — compile-verified
//
#include <hip/hip_runtime.h>

typedef __attribute__((ext_vector_type(2))) float v2f;
typedef __attribute__((ext_vector_type(4))) float v4f;
typedef __attribute__((ext_vector_type(8))) float v8f;

#define R_VAL   1.0f
#define IF_VAL  2.0e-05f
#define C_VAL   1.0e-07f
#define VT_VAL  0.026f
#define K_STEP  (1.0f / 44100.0f)

// x_n = M @ x_n1 + vv * (u_n + u_n1);  y = gain * x_n[3];  u passthrough.
// WMMA mapping (V_WMMA_F32_16X16X4_F32, D = A(16x4) x B(4x16) + C):
//   A = M zero-padded to 16 rows (uniform across the tile loop)
//   B = X1^T : B[k][n] = x_n1[elem n][state k]   (one b64 load per lane)
//   C[s][n]  = vv[s] * (u_n[n] + u_n1[n])
//   D[s][n]  = new state s of element n -> VGPR s, lane n (lanes 0-15)
__global__ __launch_bounds__(256) void moog_vcf_wmma_kernel(
    const float* __restrict__ u_n,  const float* __restrict__ x_n1,
    const float* __restrict__ u_n1,
    const float* __restrict__ gIf,  const float* __restrict__ gC,
    const float* __restrict__ gVT,  const float* __restrict__ gr,
    const float* __restrict__ gain_p,
    float* __restrict__ y_out, float* __restrict__ x_out,
    float* __restrict__ u_out, int n)
{
    // ---- uniform scalar setup: build M (4x4) and vv (4) from params ----
    float g  = (gIf[0] * IF_VAL) / (4.0f * gC[0] * C_VAL * gVT[0] * VT_VAL);
    float a  = 0.5f * K_STEP * g;            // k*g/2
    float b  = 2.0f * K_STEP * g * gr[0] * R_VAL;  // (k/2)*4*g*gr*R
    float d1 = 1.0f + a;                     // diag of (I - kA/2)
    float p  = 1.0f - a;                     // diag of (I + kA/2)
    float d2 = d1 * d1, d3 = d2 * d1;
    float delta = d2 * d2 + b * a * a * a;   // det(I - kA/2)

    // inv(I - kA/2) column-by-column via closed-form solve of the
    // bidiagonal-plus-corner system.
    float inv[4][4];
#pragma unroll
    for (int j = 0; j < 4; ++j) {
        float y0 = (j == 0) ? 1.0f : 0.0f;
        float y1 = (j == 1) ? 1.0f : 0.0f;
        float y2 = (j == 2) ? 1.0f : 0.0f;
        float y3 = (j == 3) ? 1.0f : 0.0f;
        float x0 = (y0 * d3 - b * a * a * y1 - b * a * d1 * y2 - b * d2 * y3) / delta;
        float x1 = (y1 + a * x0) / d1;
        float x2 = (y2 + a * x1) / d1;
        float x3 = (y3 + a * x2) / d1;
        inv[0][j] = x0; inv[1][j] = x1; inv[2][j] = x2; inv[3][j] = x3;
    }
    float M[4][4], vv[4];
#pragma unroll
    for (int i = 0; i < 4; ++i) {
        M[i][0] = inv[i][0] * p + inv[i][1] * a;
        M[i][1] = inv[i][1] * p + inv[i][2] * a;
        M[i][2] = inv[i][2] * p + inv[i][3] * a;
        M[i][3] = inv[i][3] * p - inv[i][0] * b;
        vv[i]   = inv[i][0] * a;
    }
    float gain = gain_p[0];

    int  lane = (int)(threadIdx.x & 31u);
    int  e    = lane & 15;
    bool lo   = lane < 16;

    // ---- A operand: M zero-padded to 16x4, per ISA 32-bit A layout ----
    // lanes 0-15: row=lane, VGPR0=K0, VGPR1=K1; lanes 16-31: row=lane-16, K2/K3
    float s0 = lo ? M[0][0] : M[0][2];
    float s1 = lo ? M[1][0] : M[1][2];
    float s2 = lo ? M[2][0] : M[2][2];
    float s3 = lo ? M[3][0] : M[3][2];
    float t0 = lo ? M[0][1] : M[0][3];
    float t1 = lo ? M[1][1] : M[1][3];
    float t2 = lo ? M[2][1] : M[2][3];
    float t3 = lo ? M[3][1] : M[3][3];
    v2f amat;
    amat.x = (e == 0) ? s0 : (e == 1) ? s1 : (e == 2) ? s2 : (e == 3) ? s3 : 0.0f;
    amat.y = (e == 0) ? t0 : (e == 1) ? t1 : (e == 2) ? t2 : (e == 3) ? t3 : 0.0f;

    int ntiles = n >> 4;
    int wave   = (int)((blockIdx.x * blockDim.x + threadIdx.x) >> 5);
    int nwaves = (int)((gridDim.x * blockDim.x) >> 5);

    for (int tile = wave; tile < ntiles; tile += nwaves) {
        int base = tile << 4;
        int elem = base + e;

        // B operand: states (k,k+1) of element `elem`, k = lo?0:2
        const float* xp = x_n1 + (((size_t)elem) << 2) + (lo ? 0 : 2);
        v2f bmat = *(const v2f*)xp;                 // coalesced b64, 256B/wave

        float uu = u_n[elem];
        float su = uu + u_n1[elem];

        v8f c;
        c[0] = vv[0] * su; c[1] = vv[1] * su;
        c[2] = vv[2] * su; c[3] = vv[3] * su;
        c[4] = 0.0f; c[5] = 0.0f; c[6] = 0.0f; c[7] = 0.0f;

        v8f dmat = __builtin_amdgcn_wmma_f32_16x16x4_f32(
            false, amat, false, bmat, (short)0, c, false, false);

        if (lo) {
            // element e's 4 new states sit in dmat[0..3] of lane e -> b128 store
            v4f xv;
            xv.x = dmat[0]; xv.y = dmat[1]; xv.z = dmat[2]; xv.w = dmat[3];
            __builtin_nontemporal_store(xv, (v4f*)(x_out + (((size_t)elem) << 2)));
            __builtin_nontemporal_store(gain * dmat[3], y_out + elem);
            __builtin_nontemporal_store(uu, u_out + elem);
        }
    }

    // ---- scalar tail for n % 16 != 0 (no WMMA; divergence-safe) ----
    int tid   = (int)(blockIdx.x * blockDim.x + threadIdx.x);
    int total = (int)(gridDim.x * blockDim.x);
    for (int i = (ntiles << 4) + tid; i < n; i += total) {
        size_t o = ((size_t)i) << 2;
        float x0 = x_n1[o + 0], x1 = x_n1[o + 1], x2 = x_n1[o + 2], x3 = x_n1[o + 3];
        float uu = u_n[i];
        float su = uu + u_n1[i];
        float r0 = M[0][0]*x0 + M[0][1]*x1 + M[0][2]*x2 + M[0][3]*x3 + vv[0]*su;
        float r1 = M[1][0]*x0 + M[1][1]*x1 + M[1][2]*x2 + M[1][3]*x3 + vv[1]*su;
        float r2 = M[2][0]*x0 + M[2][1]*x1 + M[2][2]*x2 + M[2][3]*x3 + vv[2]*su;
        float r3 = M[3][0]*x0 + M[3][1]*x1 + M[3][2]*x2 + M[3][3]*x3 + vv[3]*su;
        x_out[o + 0] = r0; x_out[o + 1] = r1; x_out[o + 2] = r2; x_out[o + 3] = r3;
        y_out[i] = gain * r3;
        u_out[i] = uu;
    }
}

extern "C" void kernel_launch(void* const* d_in, const int* in_sizes, int n_in,
                              void* d_out, int out_size, void* d_ws, size_t ws_size,
                              hipStream_t stream) {
    const float* u_n  = (const float*)d_in[0];
    const float* x_n1 = (const float*)d_in[1];
    const float* u_n1 = (const float*)d_in[2];
    const float* gIf  = (const float*)d_in[3];
    const float* gC   = (const float*)d_in[4];
    const float* gVT  = (const float*)d_in[5];
    const float* gr   = (const float*)d_in[6];
    const float* gain = (const float*)d_in[7];

    int B = in_sizes[0];  // u_n has B elements

    float* y_out = (float*)d_out;            // [B]
    float* x_out = y_out + (size_t)B;        // [B*4]
    float* u_out = y_out + (size_t)5 * B;    // [B]

    int ntiles = B >> 4;
    int blocks = (ntiles + 7) / 8;           // 8 waves (tiles) per 256-thread block
    if (blocks < 1)    blocks = 1;
    if (blocks > 4096) blocks = 4096;        // ~32K waves, grid-stride over tiles

    moog_vcf_wmma_kernel<<<blocks, 256, 0, stream>>>(
        u_n, x_n1, u_n1, gIf, gC, gVT, gr, gain, y_out, x_out, u_out, B);
}